// GcnNet_64991445123382
// MI455X (gfx1250) — compile-verified
//
#include <hip/hip_runtime.h>

// GCN forward for MI455X (gfx1250, wave32).
// Dense x@W GEMMs use V_WMMA_F32_16X16X4_F32 (exact fp32; GEMM is compute-trivial
// at ~11.5 GFLOP total). The pipeline is memory/atomic bound on the edge scatter
// (~8 GB, L2-resident given 192 MB L2: both N*H ping-pong buffers + x fit).
// Fusions: ReLU of layer L folded into layer L+1's GEMM A-load (template param,
// no per-element selects); self-loop term h*dinv^2 + bias produced directly in
// the GEMM epilogue (kills a 102 MB/layer elementwise pass).

typedef __attribute__((ext_vector_type(2))) float v2f;
typedef __attribute__((ext_vector_type(8))) float v8f;

#define CMAX 8

// ---------- degree / normalization ----------
__global__ void fill_ones_k(float* __restrict__ deg, int N) {
  int i = blockIdx.x * blockDim.x + threadIdx.x;
  if (i < N) deg[i] = 1.0f;                       // self-loop contributes +1
}

__global__ void deg_accum_k(const int* __restrict__ dst, float* __restrict__ deg, int E) {
  int i = blockIdx.x * blockDim.x + threadIdx.x;
  if (i < E) atomicAdd(&deg[dst[i]], 1.0f);
}

__global__ void to_rsqrt_k(float* __restrict__ deg, int N) {
  int i = blockIdx.x * blockDim.x + threadIdx.x;
  if (i < N) deg[i] = rsqrtf(deg[i]);             // deg -> dinv in place
}

// ---------- dense GEMM + self-loop epilogue ----------
// Out[N,Hd] = (RELU? relu(A) : A)[N,K] @ W[K,Hd]
// Acc[N,Hd] = Out * dinv^2 + bias        (scatter accumulator pre-init)
// One 256-thread block = 8 waves; block b owns M-tile b, wave w owns N-tile w.
// fp32 WMMA 16x16x4 layouts (ISA 7.12.2):
//   A 16x4 : lanes 0-15 hold K=k,k+1 (v0,v1); lanes 16-31 hold K=k+2,k+3
//   B 4x16 : same K split across half-waves, N = lane%16
//   C/D    : VGPR r -> row r (lanes 0-15) / row r+8 (lanes 16-31), N = lane%16
template <bool RELU>
__global__ __launch_bounds__(256)
void gemm_wmma_f32(const float* __restrict__ A, const float* __restrict__ W,
                   const float* __restrict__ dinv, const float* __restrict__ bias,
                   float* __restrict__ Out, float* __restrict__ Acc,
                   int N, int K, int Hd) {
  const int mtile = blockIdx.x;
  const int wave  = threadIdx.x >> 5;
  const int lane  = threadIdx.x & 31;
  const int col16 = lane & 15;
  const int hi    = lane >> 4;            // half-wave: 0 or 1
  const int ncol  = wave * 16 + col16;
  int mrow = mtile * 16 + col16;
  if (mrow > N - 1) mrow = N - 1;         // clamp loads; keeps EXEC all-ones for WMMA
  const float* arow = A + (size_t)mrow * K;
  const int kb = 2 * hi;

  v8f c = {};
#pragma unroll 8
  for (int k = 0; k < K; k += 4) {
    float a0 = arow[k + kb + 0];
    float a1 = arow[k + kb + 1];
    if (RELU) { a0 = fmaxf(a0, 0.0f); a1 = fmaxf(a1, 0.0f); }  // compile-time fused ReLU
    v2f a; a[0] = a0; a[1] = a1;
    v2f b;
    b[0] = W[(size_t)(k + kb + 0) * Hd + ncol];
    b[1] = W[(size_t)(k + kb + 1) * Hd + ncol];
    c = __builtin_amdgcn_wmma_f32_16x16x4_f32(false, a, false, b, (short)0, c,
                                              false, false);
  }

  // Fused epilogue: h tile -> Out, and self-loop+bias pre-init -> Acc.
  const int m0 = mtile * 16 + hi * 8;
  const float bval = bias[ncol];
#pragma unroll
  for (int r = 0; r < 8; ++r) {
    int m = m0 + r;
    if (m < N) {
      float dv = dinv[m];
      size_t idx = (size_t)m * Hd + ncol;
      Out[idx] = c[r];
      Acc[idx] = c[r] * dv * dv + bval;
    }
  }
}

// ---------- edge scatter: acc[dst] += h[src] * dinv[src]*dinv[dst] ----------
// One wave per edge; 128 features / 32 lanes, fully coalesced f32 atomics.
__global__ __launch_bounds__(256)
void scatter_edges_k(const float* __restrict__ h, const int* __restrict__ src,
                     const int* __restrict__ dst, const float* __restrict__ dinv,
                     float* __restrict__ acc, int E, int Hd) {
  int e = (int)(((size_t)blockIdx.x * blockDim.x + threadIdx.x) >> 5);
  int lane = threadIdx.x & 31;
  if (e >= E) return;
  int s = src[e], d = dst[e];
  float norm = dinv[s] * dinv[d];
  const float* hs = h + (size_t)s * Hd;
  float* ad = acc + (size_t)d * Hd;
  for (int f = lane; f < Hd; f += 32)
    atomicAdd(&ad[f], hs[f] * norm);
}

// ---------- pooling ----------
__global__ void pool_zero_k(float* __restrict__ sums, unsigned* __restrict__ maxs,
                            float* __restrict__ cnts, int G, int Hd) {
  int i = blockIdx.x * blockDim.x + threadIdx.x;
  int tot = G * Hd;
  if (i < tot) { sums[i] = 0.0f; maxs[i] = 0u; }  // 0 bits == 0.0f; values >=0 post-ReLU
  if (i < G) cnts[i] = 0.0f;
}

__global__ __launch_bounds__(256)
void pool_accum_k(const float* __restrict__ acc, const int* __restrict__ batch,
                  float* __restrict__ sums, unsigned* __restrict__ maxs,
                  float* __restrict__ cnts, int N, int Hd) {
  int n = (int)(((size_t)blockIdx.x * blockDim.x + threadIdx.x) >> 5);
  int lane = threadIdx.x & 31;
  if (n >= N) return;
  int g = batch[n];
  const float* row = acc + (size_t)n * Hd;
  float* srow = sums + (size_t)g * Hd;
  unsigned* mrow = maxs + (size_t)g * Hd;
  for (int f = lane; f < Hd; f += 32) {
    float v = fmaxf(row[f], 0.0f);                // fused ReLU of layer 4
    atomicAdd(&srow[f], v);
    atomicMax(&mrow[f], __float_as_uint(v));      // valid: v >= 0 -> bits monotone
  }
  if (lane == 0) atomicAdd(&cnts[g], 1.0f);
}

// ---------- classifier + log_softmax: one wave per graph ----------
__global__ void classify_k(const float* __restrict__ sums, const float* __restrict__ maxs,
                           const float* __restrict__ cnts, const float* __restrict__ Wf,
                           const float* __restrict__ bf, float* __restrict__ out,
                           int Hd, int C) {
  int g = blockIdx.x;
  int lane = threadIdx.x;                          // blockDim.x == 32
  float cnt = fmaxf(cnts[g], 1.0f);
  float accv[CMAX];
  for (int c = 0; c < CMAX; ++c) accv[c] = 0.0f;
  int twoH = 2 * Hd;
  for (int f = lane; f < twoH; f += 32) {
    float p = (f < Hd) ? sums[(size_t)g * Hd + f] / cnt
                       : maxs[(size_t)g * Hd + (f - Hd)];
    for (int c = 0; c < C; ++c) accv[c] += p * Wf[(size_t)f * C + c];
  }
  for (int off = 16; off > 0; off >>= 1)
    for (int c = 0; c < C; ++c)
      accv[c] += __shfl_xor(accv[c], off, 32);
  if (lane == 0) {
    float m = -3.402823466e38f;
    for (int c = 0; c < C; ++c) { accv[c] += bf[c]; m = fmaxf(m, accv[c]); }
    float s = 0.0f;
    for (int c = 0; c < C; ++c) s += expf(accv[c] - m);
    float lse = logf(s) + m;
    for (int c = 0; c < C; ++c) out[(size_t)g * C + c] = accv[c] - lse;
  }
}

// ---------- host ----------
extern "C" void kernel_launch(void* const* d_in, const int* in_sizes, int n_in,
                              void* d_out, int out_size, void* d_ws, size_t ws_size,
                              hipStream_t stream) {
  const float* x     = (const float*)d_in[0];
  const int*   ei    = (const int*)d_in[1];
  const int*   batch = (const int*)d_in[2];
  const float* W[4]  = {(const float*)d_in[3], (const float*)d_in[5],
                        (const float*)d_in[7], (const float*)d_in[9]};
  const float* b[4]  = {(const float*)d_in[4], (const float*)d_in[6],
                        (const float*)d_in[8], (const float*)d_in[10]};
  const float* Wf    = (const float*)d_in[11];
  const float* bf    = (const float*)d_in[12];

  const int N  = in_sizes[2];
  const int E  = in_sizes[1] / 2;
  const int Hd = in_sizes[4];        // 128
  const int Fi = in_sizes[0] / N;    // 64
  const int C  = in_sizes[12];       // 6
  const int G  = out_size / C;       // 1024
  const int* src = ei;
  const int* dst = ei + E;

  // workspace layout (256B aligned): dinv | hbuf | abuf | sums | maxs | cnts
  char* ws = (char*)d_ws;
  size_t off = 0;
  auto alloc = [&](size_t bytes) { char* p = ws + off;
                                   off = (off + bytes + 255) & ~(size_t)255; return p; };
  float*    dinv = (float*)alloc((size_t)N * 4);
  float*    hbuf = (float*)alloc((size_t)N * Hd * 4);
  float*    abuf = (float*)alloc((size_t)N * Hd * 4);
  float*    sums = (float*)alloc((size_t)G * Hd * 4);
  unsigned* maxs = (unsigned*)alloc((size_t)G * Hd * 4);
  float*    cnts = (float*)alloc((size_t)G * 4);
  (void)ws_size; (void)n_in;

  // GCN normalization
  fill_ones_k<<<(N + 255) / 256, 256, 0, stream>>>(dinv, N);
  deg_accum_k<<<(E + 255) / 256, 256, 0, stream>>>(dst, dinv, E);
  to_rsqrt_k<<<(N + 255) / 256, 256, 0, stream>>>(dinv, N);

  // 4 conv layers. GEMM writes h (scatter source) AND the pre-initialized
  // accumulator (self-loop + bias) in one pass; scatter then adds edge messages.
  const float* in = x;
  int K = Fi;
  const int mt = (N + 15) / 16;
  const int gemm_block = (Hd / 16) * 32;  // 8 waves x 32 = 256
  for (int layer = 0; layer < 4; ++layer) {
    if (layer == 0)
      gemm_wmma_f32<false><<<mt, gemm_block, 0, stream>>>(in, W[layer], dinv, b[layer],
                                                          hbuf, abuf, N, K, Hd);
    else
      gemm_wmma_f32<true><<<mt, gemm_block, 0, stream>>>(in, W[layer], dinv, b[layer],
                                                         hbuf, abuf, N, K, Hd);
    scatter_edges_k<<<(E + 7) / 8, 256, 0, stream>>>(hbuf, src, dst, dinv, abuf, E, Hd);
    in = abuf;
    K = Hd;
  }

  // pooling (ReLU fused into the read) + classifier
  pool_zero_k<<<(G * Hd + 255) / 256, 256, 0, stream>>>(sums, maxs, cnts, G, Hd);
  pool_accum_k<<<(N + 7) / 8, 256, 0, stream>>>(abuf, batch, sums, maxs, cnts, N, Hd);
  classify_k<<<G, 32, 0, stream>>>(sums, (const float*)maxs, cnts, Wf, bf,
                                   (float*)d_out, Hd, C);
}